// TextureLoss_89816356094407
// MI455X (gfx1250) — compile-verified
//
#include <hip/hip_runtime.h>
#include <cstdint>

#define TILE_W 64
#define TILE_H 16
#define HALO 2
#define LW (TILE_W + 2*HALO)        // 68
#define LH (TILE_H + 2*HALO)        // 20
#define LELEMS (LW*LH)              // 1360
#define IMG_H 512
#define IMG_W 512
#define PLANES 48                   // 16*3
#define TILES_X (IMG_W / TILE_W)    // 8
#define TILES_PER_PLANE ((IMG_W/TILE_W)*(IMG_H/TILE_H))   // 256
#define NBLOCKS (PLANES * TILES_PER_PLANE)                // 12288
#define NTOTAL (16.0*3.0*512.0*512.0)

typedef float v2f __attribute__((ext_vector_type(2)));
typedef float v8f __attribute__((ext_vector_type(8)));

// ---- CDNA5 async global->LDS copy (ASYNCcnt path, gfx1250) -----------------
__device__ __forceinline__ void async_load_to_lds_b32(uint32_t lds_off, const float* g) {
  asm volatile("global_load_async_to_lds_b32 %0, %1, off"
               :: "v"(lds_off), "v"(g)
               : "memory");
}
__device__ __forceinline__ void wait_asynccnt0() {
  asm volatile("s_wait_asynccnt 0" ::: "memory");
}

// ---- LBP 'var' at one pixel from an LDS tile (weights folded at compile time)
__device__ __forceinline__ float lbp_var_at(const float* __restrict__ t, int lr, int lc) {
  // rp = round(-2*sin(2*pi*i/16), 8), cp = round(2*cos(2*pi*i/16), 8)
  constexpr double RP[16] = { 0.0, -0.76536686, -1.41421356, -1.84775907,
                             -2.0, -1.84775907, -1.41421356, -0.76536686,
                              0.0,  0.76536686,  1.41421356,  1.84775907,
                              2.0,  1.84775907,  1.41421356,  0.76536686};
  constexpr double CP[16] = { 2.0,  1.84775907,  1.41421356,  0.76536686,
                              0.0, -0.76536686, -1.41421356, -1.84775907,
                             -2.0, -1.84775907, -1.41421356, -0.76536686,
                              0.0,  0.76536686,  1.41421356,  1.84775907};
  constexpr int FR[16] = { 0, -1, -2, -2, -2, -2, -2, -1,  0,  0,  1,  1,  2,  1,  1,  0};
  constexpr int FC[16] = { 2,  1,  1,  0,  0, -1, -2, -2, -2, -2, -2, -1,  0,  0,  1,  1};

  float s1 = 0.0f, s2 = 0.0f;
#pragma unroll
  for (int i = 0; i < 16; ++i) {
    const double tr = RP[i] - (double)FR[i];
    const double tc = CP[i] - (double)FC[i];
    const double w00 = (1.0 - tr) * (1.0 - tc);
    const double w01 = (1.0 - tr) * tc;
    const double w10 = tr * (1.0 - tc);
    const double w11 = tr * tc;
    float v = 0.0f;
    if (w00 > 1e-12) v += (float)w00 * t[(lr + FR[i]    ) * LW + (lc + FC[i]    )];
    if (w01 > 1e-12) v += (float)w01 * t[(lr + FR[i]    ) * LW + (lc + FC[i] + 1)];
    if (w10 > 1e-12) v += (float)w10 * t[(lr + FR[i] + 1) * LW + (lc + FC[i]    )];
    if (w11 > 1e-12) v += (float)w11 * t[(lr + FR[i] + 1) * LW + (lc + FC[i] + 1)];
    s1 += v;
    s2 += v * v;
  }
  const float mean = s1 * 0.0625f;
  return s2 * 0.0625f - mean * mean;
}

// ---- wave32 reduction using V_WMMA_F32_16X16X4_F32 (B = ones => row sums) --
__device__ __forceinline__ float wave_reduce_wmma(float x) {
  v2f a; a[0] = x;    a[1] = 0.0f;   // A[16x4]: lane m -> A[m,0], lane m+16 -> A[m,2]
  v2f b; b[0] = 1.0f; b[1] = 1.0f;   // B[4x16] = ones (layout-independent)
  v8f c = {};
  v8f d = __builtin_amdgcn_wmma_f32_16x16x4_f32(false, a, false, b, (short)0, c, false, false);
  float t = d[0] + d[1] + d[2] + d[3] + d[4] + d[5] + d[6] + d[7];
  // lanes 0-15 hold sum of rows 0..7, lanes 16-31 hold sum of rows 8..15
  return __shfl(t, 0, 32) + __shfl(t, 16, 32);
}

__global__ void __launch_bounds__(256)
lbpvar_mse_kernel(const float* __restrict__ A, const float* __restrict__ B,
                  float* __restrict__ part) {
  __shared__ float tA[LELEMS];
  __shared__ float tB[LELEMS];
  __shared__ float wsum[8];

  const int bid   = blockIdx.x;
  const int plane = bid / TILES_PER_PLANE;
  const int tdx   = bid % TILES_PER_PLANE;
  const int ty    = tdx / TILES_X;
  const int tx    = tdx % TILES_X;
  const int r0    = ty * TILE_H;
  const int c0    = tx * TILE_W;
  const size_t pbase = (size_t)plane * (size_t)(IMG_H * IMG_W);
  const int tid = threadIdx.x;

  // Stage both halo tiles via async global->LDS DMA; zero-fill outside image.
  for (int e = tid; e < LELEMS; e += 256) {
    const int lr = e / LW, lc = e % LW;
    const int gr = r0 + lr - HALO;
    const int gc = c0 + lc - HALO;
    if ((unsigned)gr < (unsigned)IMG_H && (unsigned)gc < (unsigned)IMG_W) {
      const size_t gi = pbase + (size_t)gr * IMG_W + (size_t)gc;
      async_load_to_lds_b32((uint32_t)(size_t)&tA[e], A + gi);
      async_load_to_lds_b32((uint32_t)(size_t)&tB[e], B + gi);
    } else {
      tA[e] = 0.0f;
      tB[e] = 0.0f;
    }
  }
  wait_asynccnt0();     // own async transfers landed in LDS
  __syncthreads();      // all waves' transfers + zero stores visible

  // 4 pixels per thread
  float acc = 0.0f;
  const int c     = tid & (TILE_W - 1);
  const int rbase = tid >> 6;            // 0..3
#pragma unroll
  for (int q = 0; q < 4; ++q) {
    const int r  = rbase + q * 4;
    const int lr = r + HALO, lc = c + HALO;
    const float va = lbp_var_at(tA, lr, lc);
    const float vb = lbp_var_at(tB, lr, lc);
    const float d  = va - vb;
    acc += d * d;
  }

  // block reduction: WMMA wave sum -> LDS -> thread 0
  const float wtot = wave_reduce_wmma(acc);
  if ((tid & 31) == 0) wsum[tid >> 5] = wtot;
  __syncthreads();
  if (tid == 0) {
    float s = 0.0f;
#pragma unroll
    for (int i = 0; i < 8; ++i) s += wsum[i];
    part[bid] = s;
  }
}

__global__ void __launch_bounds__(256)
lbpvar_reduce_kernel(const float* __restrict__ part, int n, float* __restrict__ out) {
  __shared__ double sh[256];
  double s = 0.0;
  for (int i = threadIdx.x; i < n; i += 256) s += (double)part[i];
  sh[threadIdx.x] = s;
  __syncthreads();
  for (int k = 128; k > 0; k >>= 1) {
    if (threadIdx.x < k) sh[threadIdx.x] += sh[threadIdx.x + k];
    __syncthreads();
  }
  if (threadIdx.x == 0) out[0] = (float)(sh[0] / NTOTAL);
}

extern "C" void kernel_launch(void* const* d_in, const int* in_sizes, int n_in,
                              void* d_out, int out_size, void* d_ws, size_t ws_size,
                              hipStream_t stream) {
  const float* A = (const float*)d_in[0];   // "output" image, f32 [16,3,512,512]
  const float* B = (const float*)d_in[1];   // "target" image, f32 [16,3,512,512]
  float* part = (float*)d_ws;               // NBLOCKS partial sums (48 KB)
  float* out  = (float*)d_out;              // scalar f32

  lbpvar_mse_kernel<<<dim3(NBLOCKS), dim3(256), 0, stream>>>(A, B, part);
  lbpvar_reduce_kernel<<<dim3(1), dim3(256), 0, stream>>>(part, NBLOCKS, out);
}